// Conv2DTransposeSeparableLayer_66022237274112
// MI455X (gfx1250) — compile-verified
//
#include <hip/hip_runtime.h>
#include <math.h>

// ---------------------------------------------------------------------------
// Conv2DTransposeSeparableLayer for MI455X (gfx1250, wave32, WMMA)
//
// Each 16-output tile of the 2x-upsampling 9-tap grouped transpose conv is a
// 16x16x36 GEMM with a constant band matrix A[r,3di+c] = inv[c, r+8-2di]
// (inv[c,q] = 0.5*cos(0.7*(9c+q))), executed as 9 chained
// v_wmma_f32_16x16x4_f32 ops. Bandwidth-bound (~415 MiB total traffic).
//
// Rev 3: cosf is instantiated exactly once per kernel — a zero-padded
// 3x38 weight table is built cooperatively in LDS; A-fragments are then
// unconditional ds_loads at base + compile-time-selected constants.
// ---------------------------------------------------------------------------

typedef __attribute__((ext_vector_type(2))) float v2f;
typedef __attribute__((ext_vector_type(4))) float v4f;
typedef __attribute__((ext_vector_type(8))) float v8f;

#define BATCH 16
#define H_IN  128
#define W_IN  128
#define C_IN  144
#define W_MID 256   // 2*W_IN
#define C_MID 48    // C_IN/3
#define H_OUT 256   // 2*H_IN
#define C_OUT 16    // C_MID/3

#define WTAB  114   // 3 sections of 38: P[c][q+14], q in [-14,23], nonzero q in [0,8]

// symmetric padding map: padded coord i in [0, n+5] -> source index in [0, n-1]
__device__ __forceinline__ int sym_map(int i, int n) {
    if (i < 3)     return 2 - i;         // left:  a2,a1,a0
    if (i > n + 2) return 2 * n + 2 - i; // right: a[n-1],a[n-2],a[n-3]
    return i - 3;
}

// border multiplier on padded coords: 0,1,2 -> 2,1.5,1.25 ; mirrored at right
__device__ __forceinline__ float border_f(int i, int n) {
    if (i == 0 || i == n + 5) return 2.0f;
    if (i == 1 || i == n + 4) return 1.5f;
    if (i == 2 || i == n + 3) return 1.25f;
    return 1.0f;
}

// Cooperative fill of the zero-padded weight table (single cosf instantiation).
// winv[38*c + (q+14)] = 0.5*cos(0.7*(9c+q)) for q in [0,8], else 0.
__device__ __forceinline__ void fill_wtab(float* winv, int tid) {
    if (tid < WTAB) {
        const int c = tid / 38;
        const int q = tid - 38 * c - 14;
        winv[tid] = (q >= 0 && q <= 8)
                        ? 0.5f * cosf(0.7f * (float)(9 * c + q))
                        : 0.0f;
    }
}

// Per-lane A fragments from the LDS table.
// 32-bit A 16x4 layout: lane L holds row m = L%16; VGPR0/1 hold K = koff,koff+1
// with koff = (L<16 ? 0 : 2).  For K-index kk: di=kk/3, c=kk%3, table index
// = 38*c + (m + 8 - 2*di) + 14 = m + (38*c + 22 - 2*di)  (constant per half).
__device__ __forceinline__ void make_a(v2f a[9], const float* winv, int m, bool lo) {
#pragma unroll
    for (int kc = 0; kc < 9; ++kc) {
        {   // .x element: kk = 4*kc + (lo?0:2)
            const int kL = 4 * kc, kH = 4 * kc + 2;
            const int oL = 38 * (kL % 3) + 22 - 2 * (kL / 3);
            const int oH = 38 * (kH % 3) + 22 - 2 * (kH / 3);
            a[kc].x = winv[m + (lo ? oL : oH)];
        }
        {   // .y element: kk = 4*kc + (lo?1:3)
            const int kL = 4 * kc + 1, kH = 4 * kc + 3;
            const int oL = 38 * (kL % 3) + 22 - 2 * (kL / 3);
            const int oH = 38 * (kH % 3) + 22 - 2 * (kH / 3);
            a[kc].y = winv[m + (lo ? oL : oH)];
        }
    }
}

// ---------------------------------------------------------------------------
// Pass 1: along W.  x(16,128,128,144) -> mid(16,128,256,48)
// grid = 2048 rows * 2 halves; block = 256 threads = 8 waves.
// Wave w computes j-subtile j0 = half*128 + 16*w, unrolled over g0 {0,16,32}.
// ---------------------------------------------------------------------------
__global__ __launch_bounds__(256) void pass1_w(const float* __restrict__ x,
                                               float* __restrict__ mid) {
    const int bh   = blockIdx.x >> 1;   // b*128 + h
    const int half = blockIdx.x & 1;
    const int tid  = threadIdx.x;
    const int lane = tid & 31;
    const int wv   = tid >> 5;
    const int m    = lane & 15;
    const bool lo  = lane < 16;

    const int j0b  = half * 128;
    const int i_lo = j0b / 2 + 1;       // first padded-W coord needed

    __shared__ float simg[68 * C_IN];   // 39168 B
    __shared__ float winv[WTAB];

    fill_wtab(winv, tid);

    // ---- stage: wave-per-row, row-uniform map/factor, float4 copies -------
    const float* xrow = x + (size_t)bh * (W_IN * C_IN);
    for (int r = wv; r < 68; r += 8) {
        const int    ip  = i_lo + r;                // padded coord (1..132)
        const float  f   = border_f(ip, W_IN);
        const float* src = xrow + sym_map(ip, W_IN) * C_IN;
        float*       dst = simg + r * C_IN;
        v4f d = *(const v4f*)(src + 4 * lane);      // floats [0,128)
        *(v4f*)(dst + 4 * lane) = d * f;
        if (lane < 4) {                             // floats [128,144)
            v4f d2 = *(const v4f*)(src + 128 + 4 * lane);
            *(v4f*)(dst + 128 + 4 * lane) = d2 * f;
        }
    }
    __syncthreads();

    v2f a[9];
    make_a(a, winv, m, lo);

    // ---- hoisted LDS read addresses (float indices) -----------------------
    // bf for chunk kc, group-tile gq: simg[1152*wv + 144*di + 3*m + c + 48*gq]
    const int base = 1152 * wv + 3 * m;
    int addr0[9], addr1[9];
#pragma unroll
    for (int kc = 0; kc < 9; ++kc) {
        const int kLx = 4 * kc,     kHx = 4 * kc + 2;
        const int kLy = 4 * kc + 1, kHy = 4 * kc + 3;
        const int oLx = 144 * (kLx / 3) + (kLx % 3);
        const int oHx = 144 * (kHx / 3) + (kHx % 3);
        const int oLy = 144 * (kLy / 3) + (kLy % 3);
        const int oHy = 144 * (kHy / 3) + (kHy % 3);
        addr0[kc] = base + (lo ? oLx : oHx);
        addr1[kc] = base + (lo ? oLy : oHy);
    }

    // ---- compute + store --------------------------------------------------
    const int j0   = j0b + 16 * wv;
    const int mrow = lo ? 0 : 8;        // C/D layout M offset
    float* p = mid + (size_t)bh * (W_MID * C_MID) + (size_t)(j0 + mrow) * C_MID + m;

#pragma unroll
    for (int gq = 0; gq < 3; ++gq) {    // g0 = 16*gq -> LDS offset 48*gq
        v8f acc = {};
#pragma unroll
        for (int kc = 0; kc < 9; ++kc) {
            v2f bf;
            bf.x = simg[addr0[kc] + 48 * gq];
            bf.y = simg[addr1[kc] + 48 * gq];
            acc = __builtin_amdgcn_wmma_f32_16x16x4_f32(
                false, a[kc], false, bf, (short)0, acc, false, false);
        }
#pragma unroll
        for (int v = 0; v < 8; ++v) {
            p[48 * v + 16 * gq] = acc[v];
        }
    }
}

// ---------------------------------------------------------------------------
// Pass 2: along H.  mid(16,128,256,48) -> out(16,256,256,16)
// grid = 16 b * 16 ho-tiles * 32 w-chunks = 8192; block = 256 = 8 waves,
// each wave owns one w within the 8-wide chunk.
// ---------------------------------------------------------------------------
__global__ __launch_bounds__(256) void pass2_h(const float* __restrict__ mid,
                                               float* __restrict__ out) {
    int idx = blockIdx.x;
    const int wch = idx & 31; idx >>= 5;    // w chunk (8 w each)
    const int ht  = idx & 15; idx >>= 4;    // ho tile
    const int b   = idx;                    // batch

    const int tid  = threadIdx.x;
    const int lane = tid & 31;
    const int wv   = tid >> 5;
    const int m    = lane & 15;
    const bool lo  = lane < 16;

    const int ho0 = 16 * ht;
    const int i0  = ho0 / 2 + 1;            // first padded-H coord (1..121)
    const int w0  = 8 * wch;

    __shared__ float sm[12 * 8 * C_MID];    // [di][wq][ch] = 18432 B
    __shared__ float winv[WTAB];

    fill_wtab(winv, tid);

    // ---- stage: wave-per-di; 8 w-rows x 48 ch are contiguous (384 floats) -
    const float* mb = mid + (size_t)b * (H_IN * W_MID * C_MID);
    for (int di = wv; di < 12; di += 8) {
        const int    ip  = i0 + di;
        const float  f   = border_f(ip, H_IN);
        const float* src = mb + ((size_t)sym_map(ip, H_IN) * W_MID + w0) * C_MID;
        float*       dst = sm + di * (8 * C_MID);
#pragma unroll
        for (int q = 0; q < 3; ++q) {       // 96 float4 over 32 lanes
            const int o = 4 * (lane + 32 * q);
            v4f d = *(const v4f*)(src + o);
            *(v4f*)(dst + o) = d * f;
        }
    }
    __syncthreads();

    v2f a[9];
    make_a(a, winv, m, lo);

    // bf for chunk kc: sm[384*di + 48*wv + 3*m + c]
    const int base = 48 * wv + 3 * m;
    int addr0[9], addr1[9];
#pragma unroll
    for (int kc = 0; kc < 9; ++kc) {
        const int kLx = 4 * kc,     kHx = 4 * kc + 2;
        const int kLy = 4 * kc + 1, kHy = 4 * kc + 3;
        const int oLx = 384 * (kLx / 3) + (kLx % 3);
        const int oHx = 384 * (kHx / 3) + (kHx % 3);
        const int oLy = 384 * (kLy / 3) + (kLy % 3);
        const int oHy = 384 * (kHy / 3) + (kHy % 3);
        addr0[kc] = base + (lo ? oLx : oHx);
        addr1[kc] = base + (lo ? oLy : oHy);
    }

    v8f acc = {};
#pragma unroll
    for (int kc = 0; kc < 9; ++kc) {
        v2f bf;
        bf.x = sm[addr0[kc]];
        bf.y = sm[addr1[kc]];
        acc = __builtin_amdgcn_wmma_f32_16x16x4_f32(
            false, a[kc], false, bf, (short)0, acc, false, false);
    }

    const int mrow = lo ? 0 : 8;
    float* p = out + (size_t)b * (H_OUT * W_MID * C_OUT)
                   + ((size_t)(ho0 + mrow) * W_MID + (w0 + wv)) * C_OUT + m;
#pragma unroll
    for (int v = 0; v < 8; ++v) {
        p[v * (W_MID * C_OUT)] = acc[v];    // offset v*16384 B, fits immediate
    }
}

// ---------------------------------------------------------------------------
extern "C" void kernel_launch(void* const* d_in, const int* in_sizes, int n_in,
                              void* d_out, int out_size, void* d_ws, size_t ws_size,
                              hipStream_t stream) {
    (void)in_sizes; (void)n_in; (void)out_size; (void)ws_size;
    const float* x   = (const float*)d_in[0];
    float*       out = (float*)d_out;
    float*       mid = (float*)d_ws;   // 16*128*256*48 floats = 96 MiB scratch

    pass1_w<<<BATCH * H_IN * 2, 256, 0, stream>>>(x, mid);
    pass2_h<<<BATCH * 16 * 32, 256, 0, stream>>>(mid, out);
}